// RDA_conv_81879256531183
// MI455X (gfx1250) — compile-verified
//
#include <hip/hip_runtime.h>
#include <hip/hip_bf16.h>

// RDA conv: per-sample/per-channel predicted 3x3 depthwise conv with
// sigmoid-gated residual. Memory-bound (AI ~2.5 flop/byte); WMMA used for the
// tiny kernel-predictor GEMMs (fp32 16x16x4 WMMA path on gfx1250).

typedef __attribute__((ext_vector_type(2))) float v2f;
typedef __attribute__((ext_vector_type(8))) float v8f;

#define B_   16
#define C_   64
#define H_   192
#define W_   192
#define HW_  (H_ * W_)
#define SLOPE 0.1f

// ---------------------------------------------------------------------------
// Kernel 1: kernel predictor.
//   hid = leaky_relu(z @ w1^T)      (16 x 64,  K=64)
//   ker = hid @ w2^T                (16 x 576, K=64)
// One block, 8 wave32s. V_WMMA_F32_16X16X4_F32, K stepped by 4.
// A lane layout (16x4 f32 A): lane L: m=L%16, VGPR j holds K = (L/16)*2 + j.
// B lane layout (4x16 f32 B): lane L: n=L%16, VGPR j holds K = (L/16)*2 + j.
// C/D (16x16 f32): lane L: n=L%16, VGPR j holds M = (L/16)*8 + j.
// ---------------------------------------------------------------------------
__global__ __launch_bounds__(256) void predict_ker(
    const float* __restrict__ z,    // (16, 64)
    const float* __restrict__ w1,   // (64, 64)
    const float* __restrict__ w2,   // (576, 64)
    float* __restrict__ ker_out)    // (16, 576) = [b][c*9 + tap]
{
  __shared__ float hid[16 * 64];
  const int tid  = threadIdx.x;
  const int wave = tid >> 5;
  const int lane = tid & 31;
  const int half = lane >> 4;   // 0 or 1
  const int l16  = lane & 15;

  // Phase 1: hid (16x64) — waves 0..3, one 16x16 N-tile each. Wave-uniform
  // branch keeps EXEC all-1s inside the WMMA region.
  if (wave < 4) {
    const int n = wave * 16 + l16;
    v8f acc = {};
#pragma unroll
    for (int ks = 0; ks < 16; ++ks) {
      const int kb = ks * 4 + half * 2;
      v2f a, b;
      a.x = z[l16 * 64 + kb];
      a.y = z[l16 * 64 + kb + 1];
      b.x = w1[n * 64 + kb];        // B[k][n] = w1[n][k]
      b.y = w1[n * 64 + kb + 1];
      acc = __builtin_amdgcn_wmma_f32_16x16x4_f32(
          false, a, false, b, (short)0, acc, false, false);
    }
#pragma unroll
    for (int j = 0; j < 8; ++j) {
      float v = acc[j];
      v = (v > 0.f) ? v : v * SLOPE;          // LeakyReLU(0.1)
      hid[(half * 8 + j) * 64 + n] = v;
    }
  }
  __syncthreads();

  // Phase 2: ker (16x576) — 36 N-tiles striped over all 8 waves.
  for (int tile = wave; tile < 36; tile += 8) {
    const int n = tile * 16 + l16;
    v8f acc = {};
#pragma unroll
    for (int ks = 0; ks < 16; ++ks) {
      const int kb = ks * 4 + half * 2;
      v2f a, b;
      a.x = hid[l16 * 64 + kb];
      a.y = hid[l16 * 64 + kb + 1];
      b.x = w2[n * 64 + kb];        // B[k][n] = w2[n][k]
      b.y = w2[n * 64 + kb + 1];
      acc = __builtin_amdgcn_wmma_f32_16x16x4_f32(
          false, a, false, b, (short)0, acc, false, false);
    }
#pragma unroll
    for (int j = 0; j < 8; ++j)
      ker_out[(half * 8 + j) * 576 + n] = acc[j];
  }
}

// ---------------------------------------------------------------------------
// Kernel 2: gate m[b,h,w] = sigmoid(sum_c x[b,c,h,w]*conv_w[c] + conv_b).
// One thread per pixel; loads coalesced along W; conv_w staged in LDS.
// Reads all of x once (151 MB) — this pass sets up L2 residency for pass 3.
// ---------------------------------------------------------------------------
__global__ __launch_bounds__(256) void gate_ker(
    const float* __restrict__ x,       // (16, 64, 192, 192)
    const float* __restrict__ conv_w,  // (64)
    const float* __restrict__ conv_b,  // (1)
    float* __restrict__ gate)          // (16, 192*192)
{
  __shared__ float cw[C_];
  const int tid = threadIdx.x;
  if (tid < C_) cw[tid] = conv_w[tid];
  __syncthreads();

  const int idx = blockIdx.x * 256 + tid;        // 0 .. 16*HW_-1, exact
  const int b   = idx / HW_;
  const int p   = idx - b * HW_;
  const float* xb = x + (size_t)b * C_ * HW_ + p;

  float acc = conv_b[0];
#pragma unroll 8
  for (int c = 0; c < C_; ++c)
    acc += xb[(size_t)c * HW_] * cw[c];

  gate[idx] = 1.0f / (1.0f + __expf(-acc));
}

// ---------------------------------------------------------------------------
// Kernel 3: depthwise 3x3 conv (predicted filter) + leaky_relu + gate + skip.
// blockIdx.y = (b,c) pair; blockIdx.x = 256-pixel strip. Taps in LDS.
// Neighborhood loads are row-contiguous; x should hit L2 (192 MB > 151 MB).
// ---------------------------------------------------------------------------
__global__ __launch_bounds__(256) void dwconv_ker(
    const float* __restrict__ x,     // (16*64, 192, 192)
    const float* __restrict__ ker,   // (16, 576) = [b][c*9 + tap]
    const float* __restrict__ gate,  // (16, HW_)
    float* __restrict__ out)         // (16*64, 192, 192)
{
  const int bc = blockIdx.y;        // 0..1023
  const int b  = bc >> 6;
  const int c  = bc & 63;

  __shared__ float k9[9];
  if (threadIdx.x < 9)
    k9[threadIdx.x] = ker[b * 576 + c * 9 + threadIdx.x];
  __syncthreads();

  const int p = blockIdx.x * 256 + threadIdx.x;  // 0..HW_-1, exact (144 blocks)
  const int h = p / W_;
  const int w = p - h * W_;
  const float* xc = x + (size_t)bc * HW_;

  float acc = 0.f;
#pragma unroll
  for (int dy = -1; dy <= 1; ++dy) {
    const int hh = h + dy;
#pragma unroll
    for (int dx = -1; dx <= 1; ++dx) {
      const int ww = w + dx;
      const bool ok = ((unsigned)hh < (unsigned)H_) & ((unsigned)ww < (unsigned)W_);
      const float v = ok ? xc[hh * W_ + ww] : 0.f;   // zero padding
      acc += v * k9[(dy + 1) * 3 + (dx + 1)];
    }
  }
  acc = (acc > 0.f) ? acc : acc * SLOPE;             // LeakyReLU(0.1)
  out[(size_t)bc * HW_ + p] = acc * gate[b * HW_ + p] + xc[p];
}

// ---------------------------------------------------------------------------
extern "C" void kernel_launch(void* const* d_in, const int* in_sizes, int n_in,
                              void* d_out, int out_size, void* d_ws, size_t ws_size,
                              hipStream_t stream) {
  const float* x      = (const float*)d_in[0];
  const float* z      = (const float*)d_in[1];
  const float* w1     = (const float*)d_in[2];
  const float* w2     = (const float*)d_in[3];
  const float* conv_w = (const float*)d_in[4];
  const float* conv_b = (const float*)d_in[5];
  float* out = (float*)d_out;

  // workspace layout: [ker: 16*576 floats][gate: 16*HW_ floats] ≈ 2.3 MB
  float* ws_ker  = (float*)d_ws;
  float* ws_gate = ws_ker + 16 * 576;

  predict_ker<<<1, 256, 0, stream>>>(z, w1, w2, ws_ker);
  gate_ker<<<(B_ * HW_) / 256, 256, 0, stream>>>(x, conv_w, conv_b, ws_gate);
  dwconv_ker<<<dim3(HW_ / 256, B_ * C_), 256, 0, stream>>>(x, ws_ker, ws_gate, out);
}